// Model_3813930959146
// MI455X (gfx1250) — compile-verified
//
#include <hip/hip_runtime.h>
#include <math.h>

// ---------------------------------------------------------------------------
// Problem constants (match reference)
// ---------------------------------------------------------------------------
#define RENDER 256
#define ORIG 640
#define IMG_H 480
#define IMG_W 640
#define NV 2048
#define NF 4096
#define NGROUPS (NF / 4)          // 4 faces per WMMA group
#define SIGMA 0.5f
#define INV_SIGMA 2.0f
#define LOG_EPS_CAP 13.815510558f // -log(1e-6)

typedef __attribute__((ext_vector_type(2))) float v2f;
typedef __attribute__((ext_vector_type(8))) float v8f;

// ws float layout
#define WS_XY      0                      // 2048*2
#define WS_STREAM  (WS_XY + NV * 2)       // NGROUPS*64 = 65536
#define WS_RENDER  (WS_STREAM + NGROUPS * 64)  // 65536
#define WS_PART    (WS_RENDER + RENDER * RENDER) // 1200
#define N_OUTPIX   (IMG_H * IMG_W)
#define RED_BLOCKS ((N_OUTPIX + 255) / 256)    // 1200

// ---------------------------------------------------------------------------
// Kernel 1: SE3 exp + vertex transform + projection -> xy[NV][2]
// ---------------------------------------------------------------------------
__global__ void k_transform(const float* __restrict__ dof,
                            const float* __restrict__ verts,
                            const float* __restrict__ Kmat,
                            float* __restrict__ ws) {
    int i = blockIdx.x * blockDim.x + threadIdx.x;
    if (i >= NV) return;

    float vx = dof[0], vy = dof[1], vz = dof[2];
    float wx = dof[3], wy = dof[4], wz = dof[5];
    float th2 = wx * wx + wy * wy + wz * wz;
    float th = sqrtf(th2 + 1e-30f);
    bool small = th < 1e-4f;
    float th_s = small ? 1.0f : th;
    float th2_s = small ? 1.0f : th2;
    float s = sinf(th_s), c = cosf(th_s);
    float A = small ? (1.0f - th2 / 6.0f) : (s / th_s);
    float B = small ? (0.5f - th2 / 24.0f) : ((1.0f - c) / th2_s);
    float C = small ? (1.0f / 6.0f - th2 / 120.0f) : ((th_s - s) / (th2_s * th_s));

    // Kx^2 = w w^T - th2 I
    float R00 = 1.0f + B * (wx * wx - th2);
    float R01 = -A * wz + B * wx * wy;
    float R02 =  A * wy + B * wx * wz;
    float R10 =  A * wz + B * wy * wx;
    float R11 = 1.0f + B * (wy * wy - th2);
    float R12 = -A * wx + B * wy * wz;
    float R20 = -A * wy + B * wz * wx;
    float R21 =  A * wx + B * wz * wy;
    float R22 = 1.0f + B * (wz * wz - th2);

    float V00 = 1.0f + C * (wx * wx - th2);
    float V01 = -B * wz + C * wx * wy;
    float V02 =  B * wy + C * wx * wz;
    float V10 =  B * wz + C * wy * wx;
    float V11 = 1.0f + C * (wy * wy - th2);
    float V12 = -B * wx + C * wy * wz;
    float V20 = -B * wy + C * wz * wx;
    float V21 =  B * wx + C * wz * wy;
    float V22 = 1.0f + C * (wz * wz - th2);

    float tx = V00 * vx + V01 * vy + V02 * vz;
    float ty = V10 * vx + V11 * vy + V12 * vz;
    float tz = V20 * vx + V21 * vy + V22 * vz;

    float px = verts[3 * i + 0], py = verts[3 * i + 1], pz = verts[3 * i + 2];
    float cx = R00 * px + R01 * py + R02 * pz + tx;
    float cy = R10 * px + R11 * py + R12 * pz + ty;
    float cz = R20 * px + R21 * py + R22 * pz + tz;

    float u = Kmat[0] * cx + Kmat[1] * cy + Kmat[2] * cz;
    float v = Kmat[3] * cx + Kmat[4] * cy + Kmat[5] * cz;
    float w = Kmat[6] * cx + Kmat[7] * cy + Kmat[8] * cz;
    float inv = 1.0f / w;
    const float scale = (float)RENDER / (float)ORIG;
    ws[WS_XY + 2 * i + 0] = u * inv * scale;
    ws[WS_XY + 2 * i + 1] = v * inv * scale;
}

// ---------------------------------------------------------------------------
// Kernel 2: per-face edge coefficients, packed directly into the WMMA A-operand
// lane order.  Group g holds faces 4g..4g+3 in D-rows {0-2,3-5,8-10,11-13};
// lane l<16 carries (cpx,cpy) of edge-row l, lane l>=16 carries (cc,0).
// stream[g*64 + 2*lane + {0,1}]
// ---------------------------------------------------------------------------
__global__ void k_coeff(const int* __restrict__ faces, float* __restrict__ ws) {
    int f = blockIdx.x * blockDim.x + threadIdx.x;
    if (f >= NF) return;
    const float* xy = ws + WS_XY;
    float* stream = ws + WS_STREAM;

    int i0 = faces[3 * f + 0], i1 = faces[3 * f + 1], i2 = faces[3 * f + 2];
    float p0x = xy[2 * i0], p0y = xy[2 * i0 + 1];
    float p1x = xy[2 * i1], p1y = xy[2 * i1 + 1];
    float p2x = xy[2 * i2], p2y = xy[2 * i2 + 1];

    float ex[3], ey[3], ax[3], ay[3];
    ex[0] = p1x - p0x; ey[0] = p1y - p0y; ax[0] = p0x; ay[0] = p0y;
    ex[1] = p2x - p1x; ey[1] = p2y - p1y; ax[1] = p1x; ay[1] = p1y;
    ex[2] = p0x - p2x; ey[2] = p0y - p2y; ax[2] = p2x; ay[2] = p2y;

    // area2 = cross2(e0, -e2)
    float area2 = ex[0] * (-ey[2]) - ey[0] * (-ex[2]);
    float sgn = (area2 >= 0.0f) ? 1.0f : -1.0f;

    int g = f >> 2;
    int j = f & 3;
    int M0 = (j == 0) ? 0 : (j == 1) ? 3 : (j == 2) ? 8 : 11;
    float* base = stream + (size_t)g * 64;

#pragma unroll
    for (int k = 0; k < 3; ++k) {
        float elen = sqrtf(ex[k] * ex[k] + ey[k] * ey[k]) + 1e-9f;
        float invl = sgn / elen;
        float c0 = ex[k] * ay[k] - ey[k] * ax[k];
        int M = M0 + k;
        base[2 * M + 0]        = -ey[k] * invl;   // coeff on px  (K=0)
        base[2 * M + 1]        =  ex[k] * invl;   // coeff on py  (K=1)
        base[2 * (M + 16) + 0] = -c0 * invl;      // constant     (K=2)
        base[2 * (M + 16) + 1] = 0.0f;            // K=3 unused
    }
    if (j == 0) {
        // zero dummy rows 6,7,14,15 (both lane halves)
#pragma unroll
        for (int r = 0; r < 4; ++r) {
            int M = (r < 2) ? (6 + r) : (12 + r); // 6,7,14,15
            base[2 * M + 0] = 0.0f;  base[2 * M + 1] = 0.0f;
            base[2 * (M + 16) + 0] = 0.0f; base[2 * (M + 16) + 1] = 0.0f;
        }
    }
}

// ---------------------------------------------------------------------------
// Kernel 3: WMMA rasterizer. One block per render row (256 threads = 8 waves),
// each wave owns 32 pixels as two 16-pixel tiles.  Per group: one A load,
// two v_wmma_f32_16x16x4_f32, per-lane min3 + softplus accumulation.
// ---------------------------------------------------------------------------
__device__ __forceinline__ float soft_contrib(float dmin) {
    float x = dmin * INV_SIGMA;
    // stable softplus(x) = max(x,0) + log(1+exp(-|x|))
    float sp = fmaxf(x, 0.0f) + __logf(1.0f + __expf(-fabsf(x)));
    return -fminf(sp, LOG_EPS_CAP); // == log1p(-min(sigmoid(x), 1-1e-6))
}

__global__ void __launch_bounds__(256) k_raster(float* __restrict__ ws) {
    const float2* stream = (const float2*)(ws + WS_STREAM);
    float* render = ws + WS_RENDER;

    int y = blockIdx.x;
    int tid = threadIdx.x;
    int wave = tid >> 5;
    int lane = tid & 31;
    int laneLo = lane & 15;
    bool hi = lane >= 16;
    int xbase = wave * 32;

    // B operand (4x16): lanes 0-15 -> rows K0=px,K1=py ; lanes 16-31 -> K2=1,K3=0
    v2f b0, b1;
    b0.x = hi ? 1.0f : (float)(xbase + laneLo) + 0.5f;
    b0.y = hi ? 0.0f : (float)y + 0.5f;
    b1.x = hi ? 1.0f : (float)(xbase + 16 + laneLo) + 0.5f;
    b1.y = hi ? 0.0f : (float)y + 0.5f;

    float lp0 = 0.0f, lp1 = 0.0f; // per-lane partial logp (2 faces/group each)
    v8f czero = {};

#pragma unroll 4
    for (int g = 0; g < NGROUPS; ++g) {
        float2 av = stream[(size_t)g * 32 + lane];
        v2f a; a.x = av.x; a.y = av.y;

        v8f d0 = __builtin_amdgcn_wmma_f32_16x16x4_f32(
            false, a, false, b0, (short)0, czero, false, false);
        float mA = fminf(fminf(d0[0], d0[1]), d0[2]);
        float mB = fminf(fminf(d0[3], d0[4]), d0[5]);
        lp0 += soft_contrib(mA) + soft_contrib(mB);

        v8f d1 = __builtin_amdgcn_wmma_f32_16x16x4_f32(
            false, a, false, b1, (short)0, czero, false, false);
        float mC = fminf(fminf(d1[0], d1[1]), d1[2]);
        float mD = fminf(fminf(d1[3], d1[4]), d1[5]);
        lp1 += soft_contrib(mC) + soft_contrib(mD);
    }

    // lane l (<16) has faces {0,1} partials; lane l+16 has faces {2,3} partials
    float t0 = lp0 + __shfl_xor(lp0, 16, 32);
    float t1 = lp1 + __shfl_xor(lp1, 16, 32);

    if (lane < 16) {
        float si0 = 1.0f - __expf(t0);
        float si1 = 1.0f - __expf(t1);
        render[y * RENDER + xbase + lane] = si0;
        render[y * RENDER + xbase + 16 + lane] = si1;
    }
}

// ---------------------------------------------------------------------------
// Kernel 4: nearest resize 256->640, crop 480x640, write si, block-partial SSE
// ---------------------------------------------------------------------------
__global__ void k_resize_loss(const float* __restrict__ mask,
                              float* __restrict__ out,
                              float* __restrict__ ws) {
    __shared__ float sdata[256];
    const float* render = ws + WS_RENDER;
    int i = blockIdx.x * blockDim.x + threadIdx.x;
    float diff2 = 0.0f;
    if (i < N_OUTPIX) {
        int oy = i / IMG_W;
        int ox = i - oy * IMG_W;
        const float sc = (float)RENDER / (float)ORIG; // 0.4
        int sy = (int)(((float)oy + 0.5f) * sc); sy = sy > (RENDER - 1) ? (RENDER - 1) : sy;
        int sx = (int)(((float)ox + 0.5f) * sc); sx = sx > (RENDER - 1) ? (RENDER - 1) : sx;
        float v = render[sy * RENDER + sx];
        out[1 + i] = v;
        float d = v - mask[i];
        diff2 = d * d;
    }
    sdata[threadIdx.x] = diff2;
    __syncthreads();
    for (int s = 128; s > 0; s >>= 1) {
        if (threadIdx.x < s) sdata[threadIdx.x] += sdata[threadIdx.x + s];
        __syncthreads();
    }
    if (threadIdx.x == 0) ws[WS_PART + blockIdx.x] = sdata[0];
}

// Kernel 5: deterministic final sum (fixed order, no float atomics)
__global__ void k_final(float* __restrict__ out, const float* __restrict__ ws) {
    if (threadIdx.x == 0 && blockIdx.x == 0) {
        float acc = 0.0f;
        for (int i = 0; i < RED_BLOCKS; ++i) acc += ws[WS_PART + i];
        out[0] = acc;
    }
}

// ---------------------------------------------------------------------------
extern "C" void kernel_launch(void* const* d_in, const int* in_sizes, int n_in,
                              void* d_out, int out_size, void* d_ws, size_t ws_size,
                              hipStream_t stream) {
    const float* dof   = (const float*)d_in[0];
    const float* verts = (const float*)d_in[1];
    const int*   faces = (const int*)d_in[2];
    const float* Kmat  = (const float*)d_in[3];
    const float* mask  = (const float*)d_in[4];
    float* out = (float*)d_out;
    float* ws  = (float*)d_ws;

    k_transform<<<NV / 256, 256, 0, stream>>>(dof, verts, Kmat, ws);
    k_coeff<<<NF / 256, 256, 0, stream>>>(faces, ws);
    k_raster<<<RENDER, 256, 0, stream>>>(ws);
    k_resize_loss<<<RED_BLOCKS, 256, 0, stream>>>(mask, out, ws);
    k_final<<<1, 32, 0, stream>>>(out, ws);
}